// SparseGATLayer_60120952209431
// MI455X (gfx1250) — compile-verified
//
#include <hip/hip_runtime.h>
#include <math.h>

#define N_NODES 50000
#define N_EDGES 800000
#define IN_F 128
#define OUT_F 128
#define NEG_SLOPE 0.2f
#define EPS_F 1e-10f

typedef float v2f __attribute__((ext_vector_type(2)));
typedef float v8f __attribute__((ext_vector_type(8)));

// ---------- helpers ----------

__device__ __forceinline__ void atomAddF(float* p, float v) {
    __hip_atomic_fetch_add(p, v, __ATOMIC_RELAXED, __HIP_MEMORY_SCOPE_AGENT);
}

// monotonic float <-> uint encoding for atomicMax over signed floats
__device__ __forceinline__ unsigned fenc(float f) {
    unsigned u = __float_as_uint(f);
    return (u & 0x80000000u) ? ~u : (u | 0x80000000u);
}
__device__ __forceinline__ float fdec(unsigned u) {
    return __uint_as_float((u & 0x80000000u) ? (u ^ 0x80000000u) : ~u);
}

// ---------- kernel 1: Wh = h @ W^T (fp32 WMMA), fused s1 = Wh@a1, s2 = Wh@a2 ----------
// One wave per 16-row tile. blockDim = 128 (4 waves). 3125 tiles total (50000/16 exact).
__global__ void __launch_bounds__(128)
gat_gemm_wmma(const float* __restrict__ h, const float* __restrict__ W,
              const float* __restrict__ a, float* __restrict__ Wh,
              float* __restrict__ s1, float* __restrict__ s2, int n_tiles) {
    __shared__ float tlds[4][16 * 128];   // 8KB per wave, 32KB per block

    const int lane = threadIdx.x & 31;
    const int wave = threadIdx.x >> 5;
    const int tile = blockIdx.x * 4 + wave;
    if (tile >= n_tiles) return;          // wave-uniform guard (EXEC all-1 for WMMA)

    const int row0 = tile * 16;
    const int half = lane >> 4;           // 0: lanes 0-15, 1: lanes 16-31
    const int mn   = lane & 15;           // M (for A/C rows sets) or N (for B/C cols)

    // stage 16x128 h-tile into LDS, coalesced float4
    float* tl = &tlds[wave][0];
    {
        const float4* hp4 = (const float4*)(h + row0 * IN_F);
        float4* tl4 = (float4*)tl;
        #pragma unroll
        for (int t = 0; t < 16; ++t)
            tl4[t * 32 + lane] = hp4[t * 32 + lane];
    }

    // per-lane slices of attention vectors (this lane's N column in each 16-col tile)
    float a1v[8], a2v[8];
    #pragma unroll
    for (int nt = 0; nt < 8; ++nt) {
        a1v[nt] = a[nt * 16 + mn];
        a2v[nt] = a[OUT_F + nt * 16 + mn];
    }

    v8f acc[8] = {};                       // 8 C tiles of 16x16 f32 (cols nt*16..nt*16+15)

    // K loop: 32 steps of K=4
    for (int kc = 0; kc < 32; ++kc) {
        const int col = kc * 4 + 2 * half; // this half-wave's K pair
        // A fragment: rows of h tile, 2 consecutive K values
        v2f av;
        {
            const float2 t = *(const float2*)(tl + mn * 128 + col);
            av[0] = t.x; av[1] = t.y;
        }
        #pragma unroll
        for (int nt = 0; nt < 8; ++nt) {
            const int ncol = nt * 16 + mn; // output column = W row
            v2f bv;
            {
                const float2 t = *(const float2*)(W + ncol * IN_F + col);
                bv[0] = t.x; bv[1] = t.y;
            }
            acc[nt] = __builtin_amdgcn_wmma_f32_16x16x4_f32(
                false, av, false, bv, (short)0, acc[nt], false, false);
        }
    }

    // write Wh: C layout -> lane holds column mn, VGPR j holds row j + 8*half
    #pragma unroll
    for (int nt = 0; nt < 8; ++nt) {
        #pragma unroll
        for (int j = 0; j < 8; ++j) {
            Wh[(row0 + j + 8 * half) * OUT_F + nt * 16 + mn] = acc[nt][j];
        }
    }

    // fused s1/s2: weighted column sums, then reduce across the 16 lanes of each half
    float p1[8], p2[8];
    #pragma unroll
    for (int j = 0; j < 8; ++j) { p1[j] = 0.f; p2[j] = 0.f; }
    #pragma unroll
    for (int nt = 0; nt < 8; ++nt) {
        #pragma unroll
        for (int j = 0; j < 8; ++j) {
            p1[j] = fmaf(acc[nt][j], a1v[nt], p1[j]);
            p2[j] = fmaf(acc[nt][j], a2v[nt], p2[j]);
        }
    }
    #pragma unroll
    for (int j = 0; j < 8; ++j) {
        #pragma unroll
        for (int off = 1; off < 16; off <<= 1) {
            p1[j] += __shfl_xor(p1[j], off, 32);
            p2[j] += __shfl_xor(p2[j], off, 32);
        }
    }
    if (mn == 0) {
        #pragma unroll
        for (int j = 0; j < 8; ++j) {
            s1[row0 + j + 8 * half] = p1[j];
            s2[row0 + j + 8 * half] = p2[j];
        }
    }
}

// ---------- kernel 2: e = leaky_relu(s1[row]+s2[col]); global max ----------
__global__ void __launch_bounds__(256)
gat_edge_score(const int* __restrict__ rows, const int* __restrict__ cols,
               const float* __restrict__ s1, const float* __restrict__ s2,
               float* __restrict__ e, unsigned* __restrict__ gmax, int nE) {
    __shared__ float smax[8];
    const int lane = threadIdx.x & 31;
    const int wave = threadIdx.x >> 5;

    float lmax = -INFINITY;
    for (int i = blockIdx.x * blockDim.x + threadIdx.x; i < nE;
         i += gridDim.x * blockDim.x) {
        float v = s1[rows[i]] + s2[cols[i]];
        v = (v > 0.f) ? v : NEG_SLOPE * v;
        e[i] = v;
        lmax = fmaxf(lmax, v);
    }
    #pragma unroll
    for (int off = 16; off >= 1; off >>= 1)
        lmax = fmaxf(lmax, __shfl_xor(lmax, off, 32));
    if (lane == 0) smax[wave] = lmax;
    __syncthreads();
    if (threadIdx.x == 0) {
        float m = smax[0];
        #pragma unroll
        for (int w = 1; w < 8; ++w) m = fmaxf(m, smax[w]);
        atomicMax(gmax, fenc(m));
    }
}

// ---------- kernel 3: exp_e = exp(e - gmax); sum_exp = segment_sum(exp_e, rows) ----------
__global__ void __launch_bounds__(256)
gat_edge_exp(const int* __restrict__ rows, const float* __restrict__ e,
             const unsigned* __restrict__ gmax, float* __restrict__ expe,
             float* __restrict__ sum_exp, int nE) {
    const float gm = fdec(*gmax);
    for (int i = blockIdx.x * blockDim.x + threadIdx.x; i < nE;
         i += gridDim.x * blockDim.x) {
        float ex = __expf(e[i] - gm);
        expe[i] = ex;
        atomAddF(&sum_exp[rows[i]], ex);
    }
}

// ---------- kernel 4: h_prime[row] += alpha * Wh[col]  (one wave per edge) ----------
__global__ void __launch_bounds__(256)
gat_scatter(const int* __restrict__ rows, const int* __restrict__ cols,
            const float* __restrict__ expe, const float* __restrict__ sum_exp,
            const float* __restrict__ Wh, float* __restrict__ out, int nE) {
    const int lane = threadIdx.x & 31;
    const int wave = threadIdx.x >> 5;
    const int eid = blockIdx.x * 8 + wave;
    if (eid >= nE) return;

    const int r = rows[eid];
    const int c = cols[eid];
    const float alpha = expe[eid] / (sum_exp[r] + EPS_F);

    const float4 w = ((const float4*)(Wh + c * OUT_F))[lane];
    float* dst = out + r * OUT_F + lane * 4;
    atomAddF(dst + 0, alpha * w.x);
    atomAddF(dst + 1, alpha * w.y);
    atomAddF(dst + 2, alpha * w.z);
    atomAddF(dst + 3, alpha * w.w);
}

// ---------- kernel 5: in-place ELU ----------
__global__ void __launch_bounds__(256)
gat_elu(float* __restrict__ out, int n) {
    int i = blockIdx.x * blockDim.x + threadIdx.x;
    if (i < n) {
        float x = out[i];
        out[i] = (x > 0.f) ? x : expm1f(x);
    }
}

// ---------- launch ----------
extern "C" void kernel_launch(void* const* d_in, const int* in_sizes, int n_in,
                              void* d_out, int out_size, void* d_ws, size_t ws_size,
                              hipStream_t stream) {
    const float* h  = (const float*)d_in[0];
    const int*   ei = (const int*)d_in[1];
    const float* W  = (const float*)d_in[2];
    const float* a  = (const float*)d_in[3];
    float* out = (float*)d_out;

    const int* rows = ei;
    const int* cols = ei + N_EDGES;

    // workspace layout (floats)
    float* ws = (float*)d_ws;
    float*    Wh      = ws;                                   // 6,400,000
    float*    s1      = Wh + (size_t)N_NODES * OUT_F;         //    50,000
    float*    s2      = s1 + N_NODES;                         //    50,000
    float*    e       = s2 + N_NODES;                         //   800,000
    float*    expe    = e + N_EDGES;                          //   800,000
    float*    sum_exp = expe + N_EDGES;                       //    50,000
    unsigned* gmax    = (unsigned*)(sum_exp + N_NODES);       //         1

    // zero accumulators (graph-capturable async memsets)
    hipMemsetAsync(out, 0, (size_t)N_NODES * OUT_F * sizeof(float), stream);
    hipMemsetAsync(sum_exp, 0, (size_t)N_NODES * sizeof(float), stream);
    hipMemsetAsync(gmax, 0, sizeof(unsigned), stream);  // 0 < fenc(any finite float)

    // 1) GEMM + score projections
    const int n_tiles = N_NODES / 16;                  // 3125, exact
    gat_gemm_wmma<<<(n_tiles + 3) / 4, 128, 0, stream>>>(h, W, a, Wh, s1, s2, n_tiles);

    // 2) edge scores + global max
    gat_edge_score<<<(N_EDGES + 255) / 256, 256, 0, stream>>>(rows, cols, s1, s2, e,
                                                              gmax, N_EDGES);
    // 3) exp + segment sum
    gat_edge_exp<<<(N_EDGES + 255) / 256, 256, 0, stream>>>(rows, e, gmax, expe,
                                                            sum_exp, N_EDGES);
    // 4) scatter-accumulate h_prime (one wave per edge)
    gat_scatter<<<(N_EDGES + 7) / 8, 256, 0, stream>>>(rows, cols, expe, sum_exp,
                                                       Wh, out, N_EDGES);
    // 5) ELU
    const int total = N_NODES * OUT_F;
    gat_elu<<<(total + 255) / 256, 256, 0, stream>>>(out, total);
}